// LengthRegulator_13099650253168
// MI455X (gfx1250) — compile-verified
//
#include <hip/hip_runtime.h>
#include <stdint.h>

namespace {
constexpr int kB = 32;
constexpr int kT = 512;
constexpr int kD = 384;               // floats per frame -> 1536 bytes
constexpr int kDurMax = 8;
constexpr int kTOut = kT * kDurMax;   // 4096 frames per batch row
}

// ---------------------------------------------------------------------------
// K1: per-batch-row duration sums (needed for the "all durations zero" edge
// case in the reference).
// ---------------------------------------------------------------------------
__global__ void lr_rowsum_kernel(const int* __restrict__ dur,
                                 int* __restrict__ rowsum) {
  __shared__ int sm[kT];
  const int b = blockIdx.x;
  const int tid = threadIdx.x;
  sm[tid] = dur[b * kT + tid];
  __syncthreads();
  for (int off = kT / 2; off > 0; off >>= 1) {
    if (tid < off) sm[tid] += sm[tid + off];
    __syncthreads();
  }
  if (tid == 0) rowsum[b] = sm[0];
}

// ---------------------------------------------------------------------------
// K2: per-row inclusive scan of (possibly edge-case-adjusted) durations.
// Writes cum[b][0..511] to workspace.
// ---------------------------------------------------------------------------
__global__ void lr_scan_kernel(const int* __restrict__ dur,
                               const int* __restrict__ rowsum,
                               int* __restrict__ cum) {
  __shared__ int sm[kT];
  __shared__ int total_sh;
  const int b = blockIdx.x;
  const int tid = threadIdx.x;
  if (tid == 0) {
    int tot = 0;
    for (int i = 0; i < kB; ++i) tot += rowsum[i];
    total_sh = tot;
  }
  int d = dur[b * kT + tid];
  __syncthreads();
  if (total_sh == 0) d = 1;   // reference: total==0 implies every row_sum==0
  sm[tid] = d;
  __syncthreads();
  // Hillis-Steele inclusive scan over 512 elements.
  for (int off = 1; off < kT; off <<= 1) {
    int v = (tid >= off) ? sm[tid - off] : 0;
    __syncthreads();
    sm[tid] += v;
    __syncthreads();
  }
  cum[b * kT + tid] = sm[tid];
}

// ---------------------------------------------------------------------------
// K3: expansion copy. One wave32 per output frame (1536 B). Valid frames are
// streamed global -> LDS -> global with the gfx1250 async DMA instructions
// (ASYNCcnt-tracked, no data VGPRs); pad frames get float4 zero stores.
// Block = 256 threads = 8 frames, all from one batch row (512 blocks/row).
// ---------------------------------------------------------------------------
__global__ void __launch_bounds__(256)
lr_expand_kernel(const float* __restrict__ xs,
                 const int* __restrict__ cum,
                 float* __restrict__ out) {
  __shared__ int scum[kT];                        // 2 KB: this row's cumsum
  __shared__ __align__(16) float stage[8 * kD];   // 12 KB: 1536 B per wave
  const int tid = threadIdx.x;
  const int lane = tid & 31;
  const int wave = tid >> 5;
  const int b = blockIdx.x >> 9;                  // 512 blocks per batch row
  const int fbase = (blockIdx.x & 511) * 8;

  const int* cumrow = cum + b * kT;
  scum[tid] = cumrow[tid];
  scum[tid + 256] = cumrow[tid + 256];
  __syncthreads();

  // Frame index within the row; wave-uniform -> make it scalar.
  const int t = __builtin_amdgcn_readfirstlane(fbase + wave);

  // searchsorted(cum, t, side='right'): first j with cum[j] > t.
  int lo = 0, hi = kT;
  while (lo < hi) {
    int mid = (lo + hi) >> 1;
    if (scum[mid] <= t) lo = mid + 1; else hi = mid;
  }
  const int total_frames = scum[kT - 1];

  float* dst = out + ((size_t)b * kTOut + t) * kD;

  if (t >= total_frames) {
    // PAD_VALUE = 0.0f, 3 x float4 per lane covers 1536 B per wave.
    float4 z = make_float4(0.f, 0.f, 0.f, 0.f);
    float4* dv = reinterpret_cast<float4*>(dst);
    dv[lane] = z;
    dv[lane + 32] = z;
    dv[lane + 64] = z;
    return;
  }

  int idx = (lo < kT - 1) ? lo : (kT - 1);
  idx = __builtin_amdgcn_readfirstlane(idx);
  const float* src = xs + ((size_t)b * kT + idx) * kD;

  // Wave-private LDS staging region; instruction offset applies to both the
  // LDS and global addresses, so one LDS-addr VGPR + one voffset VGPR cover
  // all three 512 B chunks.
  const uint32_t lds_addr =
      (uint32_t)(uintptr_t)(&stage[wave * kD]) + (uint32_t)(lane * 16);
  const uint32_t voff = (uint32_t)(lane * 16);

  asm volatile(
      "global_load_async_to_lds_b128 %0, %1, %2 offset:0\n\t"
      "global_load_async_to_lds_b128 %0, %1, %2 offset:512\n\t"
      "global_load_async_to_lds_b128 %0, %1, %2 offset:1024\n\t"
      "s_wait_asynccnt 0"
      :
      : "v"(lds_addr), "v"(voff), "s"(src)
      : "memory");

  asm volatile(
      "global_store_async_from_lds_b128 %1, %0, %2 offset:0\n\t"
      "global_store_async_from_lds_b128 %1, %0, %2 offset:512\n\t"
      "global_store_async_from_lds_b128 %1, %0, %2 offset:1024"
      :
      : "v"(lds_addr), "v"(voff), "s"(dst)
      : "memory");
  // s_endpgm performs an implicit wait-idle, draining ASYNCcnt.
}

// ---------------------------------------------------------------------------
extern "C" void kernel_launch(void* const* d_in, const int* in_sizes, int n_in,
                              void* d_out, int out_size, void* d_ws,
                              size_t ws_size, hipStream_t stream) {
  const float* xs = (const float*)d_in[0];   // [32, 512, 384] f32
  const int* dur = (const int*)d_in[1];      // [32, 512] i32
  float* out = (float*)d_out;                // [32, 4096, 384] f32

  int* rowsum = (int*)d_ws;                        // 32 ints
  int* cum = (int*)((char*)d_ws + 128);            // 32*512 ints = 64 KB

  lr_rowsum_kernel<<<kB, kT, 0, stream>>>(dur, rowsum);
  lr_scan_kernel<<<kB, kT, 0, stream>>>(dur, rowsum, cum);
  lr_expand_kernel<<<kB * (kTOut / 8), 256, 0, stream>>>(xs, cum, out);
}